// MeshInterpolator_11690900980008
// MI455X (gfx1250) — compile-verified
//
#include <hip/hip_runtime.h>

typedef __attribute__((ext_vector_type(2))) float v2f;
typedef __attribute__((ext_vector_type(8))) float v8f;

#define NMESH 128
#define NCH 8
#define MESH_FLOATS (NMESH * NMESH * NMESH * NCH)   // 16,777,216 floats = 64 MB
#define MESH_BYTES ((size_t)MESH_FLOATS * 4)

// ---------------------------------------------------------------------------
// Kernel 0: zero the 64MB mesh with 128-bit stores
// ---------------------------------------------------------------------------
__global__ void zero_mesh_kernel(float4* __restrict__ mesh, int n4) {
    int i = blockIdx.x * blockDim.x + threadIdx.x;
    int stride = gridDim.x * blockDim.x;
    float4 z = {0.0f, 0.0f, 0.0f, 0.0f};
    for (; i < n4; i += stride) mesh[i] = z;
}

// ---------------------------------------------------------------------------
// Kernel 1: pos_rel = ns * (positions @ inv_cell) via V_WMMA_F32_16X16X4_F32.
// One wave handles 16 particles: A(16x4) = particle coords (K = x,y,z,0),
// B(4x16) = inv_cell * 128 (shared across all particles), D(16x16) columns
// 0..2 hold pos_rel.
//
// Operand layouts (CDNA5 ISA 7.12.2, 32-bit):
//   A 16x4 : lanes 0-15 M=0..15 {V0=K0, V1=K1}; lanes 16-31 {V0=K2, V1=K3}
//   B 4x16 : lanes 0-15 N=0..15 {V0=K0 row, V1=K1 row}; lanes 16-31 {K2, K3}
//   D 16x16: lane L, VGPR v -> element (M = v + (L<16?0:8), N = L&15)
// ---------------------------------------------------------------------------
__global__ void posrel_wmma_kernel(const float* __restrict__ pos,
                                   const float* __restrict__ cell,
                                   float* __restrict__ posrel, int P) {
    const int lane = threadIdx.x & 31;
    const int wave = threadIdx.x >> 5;
    const int pbase = (blockIdx.x * (blockDim.x >> 5) + wave) * 16;
    const int n = lane & 15;          // N index / particle-within-wave
    const bool hi = lane >= 16;

    // inv(cell) * 128 (ns folded in; ns is uniform 128 per axis)
    float a = cell[0], b = cell[1], c0 = cell[2];
    float d = cell[3], e = cell[4], f = cell[5];
    float g = cell[6], h = cell[7], i9 = cell[8];
    float A0 = e * i9 - f * h;
    float B0 = -(d * i9 - f * g);
    float C0 = d * h - e * g;
    float det = a * A0 + b * B0 + c0 * C0;
    float r = 128.0f / det;
    float T00 = A0 * r,              T01 = -(b * i9 - c0 * h) * r, T02 = (b * f - c0 * e) * r;
    float T10 = B0 * r,              T11 = (a * i9 - c0 * g) * r,  T12 = -(a * f - c0 * d) * r;
    float T20 = C0 * r,              T21 = -(a * h - b * g) * r,   T22 = (a * e - b * d) * r;

    // A operand: particle coordinates (clamped load so EXEC stays full)
    int p = pbase + n;
    if (p >= P) p = P - 1;
    float px = pos[p * 3 + 0], py = pos[p * 3 + 1], pz = pos[p * 3 + 2];
    v2f Aop;
    Aop.x = hi ? pz : px;     // K2 : K0
    Aop.y = hi ? 0.0f : py;   // K3 : K1

    // B operand: rows of T at column n (zero for n >= 3), select-based (no branches)
    float r0 = (n == 0) ? T00 : ((n == 1) ? T01 : ((n == 2) ? T02 : 0.0f));
    float r1 = (n == 0) ? T10 : ((n == 1) ? T11 : ((n == 2) ? T12 : 0.0f));
    float r2 = (n == 0) ? T20 : ((n == 1) ? T21 : ((n == 2) ? T22 : 0.0f));
    v2f Bop;
    Bop.x = hi ? r2 : r0;     // K2 row : K0 row
    Bop.y = hi ? 0.0f : r1;   // K3 row : K1 row

    v8f Cop = {};
    v8f D = __builtin_amdgcn_wmma_f32_16x16x4_f32(false, Aop, false, Bop,
                                                  (short)0, Cop, false, false);

    // Extract: lane with N = n < 3 holds pos_rel component n of particles
    // pbase + v (+8 for high lanes), v = VGPR index.
    if (n < 3) {
#pragma unroll
        for (int v = 0; v < 8; ++v) {
            int pm = pbase + v + (hi ? 8 : 0);
            if (pm < P) posrel[pm * 3 + n] = D[v];
        }
    }
}

// ---------------------------------------------------------------------------
// Per-axis P3M order-4 weights + periodic indices
// ---------------------------------------------------------------------------
__device__ __forceinline__ void p3m_weights(const float pr[3], int idx[3][4],
                                            float w[3][4]) {
#pragma unroll
    for (int dd = 0; dd < 3; ++dd) {
        float fl = floorf(pr[dd]);
        float x = pr[dd] - fl - 0.5f;
        int i0 = (int)fl;
        float x2 = x * x, x3 = x2 * x;
        const float cc = 1.0f / 48.0f;
        w[dd][0] = cc * (1.0f - 6.0f * x + 12.0f * x2 - 8.0f * x3);
        w[dd][1] = cc * (23.0f - 30.0f * x - 12.0f * x2 + 24.0f * x3);
        w[dd][2] = cc * (23.0f + 30.0f * x - 12.0f * x2 - 24.0f * x3);
        w[dd][3] = cc * (1.0f + 6.0f * x + 12.0f * x2 + 8.0f * x3);
#pragma unroll
        for (int s = 0; s < 4; ++s)
            idx[dd][s] = (i0 + s - 1 + NMESH) & (NMESH - 1);
    }
}

// ---------------------------------------------------------------------------
// Kernel 2: scatter — lane = (particle, channel); 8 lanes of a particle hit
// 8 consecutive floats per node (one 32B transaction). Mesh is L2-resident.
// ---------------------------------------------------------------------------
__global__ void scatter_kernel(const float* __restrict__ posrel,
                               const float* __restrict__ pw,
                               float* __restrict__ mesh, int P) {
    int tid = blockIdx.x * blockDim.x + threadIdx.x;
    int p = tid >> 3;
    if (p >= P) return;
    int ch = tid & 7;
    float q = pw[tid];  // pw[p*8 + ch]

    float pr[3] = {posrel[p * 3 + 0], posrel[p * 3 + 1], posrel[p * 3 + 2]};
    int idx[3][4];
    float w[3][4];
    p3m_weights(pr, idx, w);

#pragma unroll
    for (int i = 0; i < 4; ++i) {
        unsigned xo = (unsigned)idx[0][i] << 7;
#pragma unroll
        for (int j = 0; j < 4; ++j) {
            unsigned xy = (xo + (unsigned)idx[1][j]) << 7;
            float wij = w[0][i] * w[1][j] * q;
#pragma unroll
            for (int k = 0; k < 4; ++k) {
                unsigned off = (((xy + (unsigned)idx[2][k]) << 3) + (unsigned)ch);
                atomicAdd(mesh + off, wij * w[2][k]);
            }
        }
    }
}

// ---------------------------------------------------------------------------
// Kernel 3: gather — same mapping, 64 coalesced L2 loads per lane
// ---------------------------------------------------------------------------
__global__ void gather_kernel(const float* __restrict__ posrel,
                              const float* __restrict__ mesh,
                              float* __restrict__ out, int P) {
    int tid = blockIdx.x * blockDim.x + threadIdx.x;
    int p = tid >> 3;
    if (p >= P) return;
    int ch = tid & 7;

    float pr[3] = {posrel[p * 3 + 0], posrel[p * 3 + 1], posrel[p * 3 + 2]};
    int idx[3][4];
    float w[3][4];
    p3m_weights(pr, idx, w);

    float sum = 0.0f;
#pragma unroll
    for (int i = 0; i < 4; ++i) {
        unsigned xo = (unsigned)idx[0][i] << 7;
#pragma unroll
        for (int j = 0; j < 4; ++j) {
            unsigned xy = (xo + (unsigned)idx[1][j]) << 7;
            float wij = w[0][i] * w[1][j];
#pragma unroll
            for (int k = 0; k < 4; ++k) {
                unsigned off = (((xy + (unsigned)idx[2][k]) << 3) + (unsigned)ch);
                sum += wij * w[2][k] * mesh[off];
            }
        }
    }
    out[tid] = sum;
}

// ---------------------------------------------------------------------------
extern "C" void kernel_launch(void* const* d_in, const int* in_sizes, int n_in,
                              void* d_out, int out_size, void* d_ws, size_t ws_size,
                              hipStream_t stream) {
    const float* positions = (const float*)d_in[0];  // (P, 3)
    const float* pweights  = (const float*)d_in[1];  // (P, 8)
    const float* cell      = (const float*)d_in[2];  // (3, 3)
    float* out = (float*)d_out;                      // (P, 8)
    int P = in_sizes[0] / 3;

    float* mesh   = (float*)d_ws;                         // 64 MB
    float* posrel = (float*)((char*)d_ws + MESH_BYTES);   // P*3 floats

    // 1) zero mesh (L2-resident accumulator)
    zero_mesh_kernel<<<4096, 256, 0, stream>>>((float4*)mesh, MESH_FLOATS / 4);

    // 2) pos_rel via WMMA (16 particles per wave, 8 waves per block)
    int nwaves = (P + 15) / 16;
    int nblk_w = (nwaves + 7) / 8;
    posrel_wmma_kernel<<<nblk_w, 256, 0, stream>>>(positions, cell, posrel, P);

    // 3) scatter + 4) gather: one lane per (particle, channel)
    int nt = P * NCH;
    int nblk = (nt + 255) / 256;
    scatter_kernel<<<nblk, 256, 0, stream>>>(posrel, pweights, mesh, P);
    gather_kernel<<<nblk, 256, 0, stream>>>(posrel, mesh, out, P);
}